// StableHierarchicalPooling_21294447854186
// MI455X (gfx1250) — compile-verified
//
#include <hip/hip_runtime.h>
#include <hip/hip_bf16.h>
#include <math.h>

typedef __attribute__((ext_vector_type(2))) float    v2f;
typedef __attribute__((ext_vector_type(8))) float    v8f;
typedef __attribute__((ext_vector_type(4))) unsigned uint32x4;
typedef __attribute__((ext_vector_type(4))) int      int32x4;
typedef __attribute__((ext_vector_type(8))) int      int32x8;

#define C_DIM 128
#define K_DIM 32
#define POOL_CHUNK 512

// padded LDS pitches (DWORDs) -> bank-conflict-free fragment reads (64 banks)
#define XP  132   // x/h tiles: pad 4 dwords every 128
#define W1P 144   // W1: pad 16 dwords every 128
#define W2P 40    // W2: pad 8 dwords every 32

__device__ __forceinline__ v8f vzero8() {
    v8f z = {0.f, 0.f, 0.f, 0.f, 0.f, 0.f, 0.f, 0.f};
    return z;
}

__device__ __forceinline__ v8f wmma_f32_4(v2f a, v2f b, v8f c) {
    // V_WMMA_F32_16X16X4_F32 : D = A(16x4) * B(4x16) + C(16x16)
    return __builtin_amdgcn_wmma_f32_16x16x4_f32(
        false, a, false, b, (short)0, c, false, false);
}

// Generic pointer to __shared__: low 32 bits are the LDS byte offset
// (ISA 10.2: LDS aperture).
__device__ __forceinline__ unsigned lds_off(const void* p) {
    return (unsigned)(unsigned long long)p;
}

// ---------------------------------------------------------------------------
// TDM: async 2D tile load Global -> LDS (D# per ISA 8.3/8.4).
// w          : tile/tensor width in f32 elements
// trows      : tile height (rows copied)
// tensor_rows: tensor height (OOB rows read as zero)
// stride     : row stride in elements
// padi/pada  : LDS pad_interval / pad_amount codes (pada < 0 -> no padding)
// ---------------------------------------------------------------------------
__device__ __forceinline__ void
tdm_load_2d(unsigned lds, const void* gptr, unsigned w, unsigned trows,
            unsigned tensor_rows, unsigned stride, int padi, int pada) {
    unsigned long long ga = (unsigned long long)gptr;
    uint32x4 g0;
    g0.x = 1u;                                                // count=1 (valid)
    g0.y = lds;                                               // lds_addr
    g0.z = (unsigned)ga;                                      // global_addr lo
    g0.w = (unsigned)((ga >> 32) & 0x01FFFFFFu) | (2u << 30); // addr hi | type=2
    unsigned w0 = (2u << 16);                                 // data_size = 4B
    if (pada >= 0)
        w0 |= (1u << 20) | ((unsigned)padi << 22) | ((unsigned)pada << 25);
    int32x8 g1;
    g1[0] = (int)w0;
    g1[1] = (int)((w & 0xFFFFu) << 16);                               // dim0 lo16
    g1[2] = (int)(((w >> 16) & 0xFFFFu) | ((tensor_rows & 0xFFFFu) << 16));
    g1[3] = (int)(((tensor_rows >> 16) & 0xFFFFu) | ((w & 0xFFFFu) << 16)); // tile_dim0
    g1[4] = (int)(trows & 0xFFFFu);                                   // tile_dim1
    g1[5] = (int)stride;                                              // dim0_stride
    g1[6] = 0;
    g1[7] = 0;
    int32x4 gz = {0, 0, 0, 0};
#if defined(__clang_major__) && (__clang_major__ >= 23)
    int32x8 gz8 = {0, 0, 0, 0, 0, 0, 0, 0};
    __builtin_amdgcn_tensor_load_to_lds(g0, g1, gz, gz, gz8, 0);
#else
    __builtin_amdgcn_tensor_load_to_lds(g0, g1, gz, gz, 0);
#endif
}

// ---------------------------------------------------------------------------
// Kernel 0: zero d_out (poisoned by harness) and v potentials.
// ---------------------------------------------------------------------------
__global__ void k_init(float* __restrict__ out, int out_n, float* __restrict__ v) {
    int i = blockIdx.x * blockDim.x + threadIdx.x;
    if (i < out_n) out[i] = 0.f;
    if (i < K_DIM) v[i] = 0.f;
}

// ---------------------------------------------------------------------------
// Kernel 1: logits = (relu(x@W1 + b1) @ W2 + b2) * scaling via f32 WMMA.
// 4 waves/block, wave-per-16-row tile. TDM stages x tiles + W1 + W2 into
// LDS (padded pitches -> conflict-free ds reads); one s_wait_tensorcnt +
// barrier, then pure WMMA from LDS.
// ---------------------------------------------------------------------------
__global__ void __launch_bounds__(128)
k_logits(const float* __restrict__ x, const float* __restrict__ W1,
         const float* __restrict__ b1, const float* __restrict__ W2,
         const float* __restrict__ b2, const float* __restrict__ scal,
         float* __restrict__ logits, int N) {
    __shared__ float xbuf[4][16 * XP];    // 33 KB
    __shared__ float hbuf[4][16 * XP];    // 33 KB
    __shared__ float w1buf[C_DIM * W1P];  // 72 KB
    __shared__ float w2buf[C_DIM * W2P];  // 20 KB

    const int tid  = threadIdx.x;
    const int wav  = tid >> 5;
    const int lane = tid & 31;
    const int half = lane >> 4;
    const int l16  = lane & 15;
    const int row0 = blockIdx.x * 64 + wav * 16;
    const float sc = scal[0];

    // --- async staging via Tensor Data Mover ---
    if (wav == 0)   // W1 [128x128], LDS pitch 144 (pad 16 per 128)
        tdm_load_2d(lds_off(w1buf), W1, C_DIM, C_DIM, C_DIM, C_DIM, 6, 15);
    if (wav == 1)   // W2 [128x32], LDS pitch 40 (pad 8 per 32)
        tdm_load_2d(lds_off(w2buf), W2, K_DIM, C_DIM, C_DIM, K_DIM, 4, 7);
    {               // own x tile [16x128], LDS pitch 132 (pad 4 per 128)
        unsigned rows = (row0 < N) ? (unsigned)(N - row0) : 0u;
        tdm_load_2d(lds_off(&xbuf[wav][0]), x + (size_t)row0 * C_DIM,
                    C_DIM, 16, rows, C_DIM, 6, 3);
    }
    __builtin_amdgcn_s_wait_tensorcnt(0);  // own TDM(s) done
    __syncthreads();                       // W1/W2 visible to all waves

    const float* xw = &xbuf[wav][0];
    float*       hw = &hbuf[wav][0];

    // ---- GEMM 1: h = relu(x @ W1 + b1), tile [16 x 128] ----
    for (int nt = 0; nt < 8; ++nt) {
        const int n0 = nt * 16;
        v8f acc = vzero8();
        #pragma unroll 4
        for (int k0 = 0; k0 < C_DIM; k0 += 4) {
            v2f a;
            a.x = xw[l16 * XP + k0 + half * 2 + 0];
            a.y = xw[l16 * XP + k0 + half * 2 + 1];
            v2f b;
            b.x = w1buf[(k0 + half * 2 + 0) * W1P + n0 + l16];
            b.y = w1buf[(k0 + half * 2 + 1) * W1P + n0 + l16];
            acc = wmma_f32_4(a, b, acc);
        }
        const int   col  = n0 + l16;
        const float bias = b1[col];
        #pragma unroll
        for (int j = 0; j < 8; ++j) {
            int r = j + half * 8;
            hw[r * XP + col] = fmaxf(acc[j] + bias, 0.f);
        }
    }
    // hbuf is wave-private; same-wave DS ordering suffices (no barrier).

    // ---- GEMM 2: logits = (h @ W2 + b2) * scaling, tile [16 x 32] ----
    for (int kt = 0; kt < 2; ++kt) {
        v8f acc = vzero8();
        #pragma unroll 4
        for (int k0 = 0; k0 < C_DIM; k0 += 4) {
            v2f a;
            a.x = hw[l16 * XP + k0 + half * 2 + 0];
            a.y = hw[l16 * XP + k0 + half * 2 + 1];
            v2f b;
            b.x = w2buf[(k0 + half * 2 + 0) * W2P + kt * 16 + l16];
            b.y = w2buf[(k0 + half * 2 + 1) * W2P + kt * 16 + l16];
            acc = wmma_f32_4(a, b, acc);
        }
        const int   col = kt * 16 + l16;
        const float bb  = b2[col];
        #pragma unroll
        for (int j = 0; j < 8; ++j) {
            int r = row0 + j + half * 8;
            if (r < N) logits[(size_t)r * K_DIM + col] = (acc[j] + bb) * sc;
        }
    }
}

// ---------------------------------------------------------------------------
// Kernel 2a: one Sinkhorn iteration, row step + per-block column partials.
//   u[n] = -LSE_k(logits[n,:] + v[:]);  partials of (logits + u + v) per column.
// Logits are L2-resident (25.6 MB << 192 MB L2).
// ---------------------------------------------------------------------------
__global__ void __launch_bounds__(256)
k_sinkrow(const float* __restrict__ logits, const float* __restrict__ v,
          float* __restrict__ u, float* __restrict__ pmax,
          float* __restrict__ psum, int N) {
    __shared__ float vs[K_DIM];
    __shared__ float wmax[8][K_DIM];
    __shared__ float wsum[8][K_DIM];

    const int tid  = threadIdx.x;
    const int wave = tid >> 5;
    const int lane = tid & 31;
    if (tid < K_DIM) vs[tid] = v[tid];
    __syncthreads();

    const int  n     = blockIdx.x * 256 + tid;
    const bool valid = (n < N);

    float z[K_DIM];
    if (valid) {
        const float4* lr = reinterpret_cast<const float4*>(logits + (size_t)n * K_DIM);
        #pragma unroll
        for (int q = 0; q < 8; ++q) {
            float4 t = lr[q];
            z[4 * q + 0] = t.x + vs[4 * q + 0];
            z[4 * q + 1] = t.y + vs[4 * q + 1];
            z[4 * q + 2] = t.z + vs[4 * q + 2];
            z[4 * q + 3] = t.w + vs[4 * q + 3];
        }
        float m = z[0];
        #pragma unroll
        for (int k = 1; k < K_DIM; ++k) m = fmaxf(m, z[k]);
        float s = 0.f;
        #pragma unroll
        for (int k = 0; k < K_DIM; ++k) s += __expf(z[k] - m);
        float un = -(m + logf(s));
        u[n] = un;
        #pragma unroll
        for (int k = 0; k < K_DIM; ++k) z[k] += un;
    } else {
        #pragma unroll
        for (int k = 0; k < K_DIM; ++k) z[k] = -1e30f;
    }

    // wave32 reduction per column k: stable (max, sumexp)
    #pragma unroll 4
    for (int k = 0; k < K_DIM; ++k) {
        float mk = z[k];
        #pragma unroll
        for (int off = 16; off > 0; off >>= 1) mk = fmaxf(mk, __shfl_xor(mk, off, 32));
        float e = __expf(z[k] - mk);
        #pragma unroll
        for (int off = 16; off > 0; off >>= 1) e += __shfl_xor(e, off, 32);
        if (lane == 0) { wmax[wave][k] = mk; wsum[wave][k] = e; }
    }
    __syncthreads();

    if (tid < K_DIM) {
        float M = -1e30f, S = 0.f;
        #pragma unroll
        for (int w = 0; w < 8; ++w) {
            float m2 = wmax[w][tid], s2 = wsum[w][tid];
            float nM = fmaxf(M, m2);
            S = S * __expf(M - nM) + s2 * __expf(m2 - nM);
            M = nM;
        }
        pmax[blockIdx.x * K_DIM + tid] = M;
        psum[blockIdx.x * K_DIM + tid] = S;
    }
}

// ---------------------------------------------------------------------------
// Kernel 2b: fold block partials -> v[k] = logT - LSE_n(logits + u + v_old)
// ---------------------------------------------------------------------------
__global__ void __launch_bounds__(32)
k_sinkcol(const float* __restrict__ pmax, const float* __restrict__ psum,
          float* __restrict__ v, int nblk, float logT) {
    const int k = threadIdx.x;
    float M = -1e30f, S = 0.f;
    for (int b = 0; b < nblk; ++b) {
        float m2 = pmax[b * K_DIM + k], s2 = psum[b * K_DIM + k];
        float nM = fmaxf(M, m2);
        S = S * __expf(M - nM) + s2 * __expf(m2 - nM);
        M = nM;
    }
    v[k] = logT - (M + logf(S));
}

// ---------------------------------------------------------------------------
// Kernel 3: out[b,k,c] += sum_n s[n,k]*x[n,c],  s = exp(logits+u+v), masked
// by batch. WMMA: M=k(32), N=c(128), reduction over n in steps of 4.
// Block = 4 waves; wave w owns c-tiles {2w,2w+1} x k-tiles {0,1}.
// ---------------------------------------------------------------------------
__global__ void __launch_bounds__(128)
k_pool(const float* __restrict__ x, const int* __restrict__ batch,
       const float* __restrict__ logits, const float* __restrict__ u,
       const float* __restrict__ v, float* __restrict__ out, int N) {
    __shared__ float vbuf[K_DIM];
    __shared__ float sbuf[POOL_CHUNK * K_DIM];   // 64 KB
    __shared__ int   bbuf[POOL_CHUNK];

    const int tid  = threadIdx.x;
    const int wav  = tid >> 5;
    const int lane = tid & 31;
    const int half = lane >> 4;
    const int l16  = lane & 15;
    const int base = blockIdx.x * POOL_CHUNK;

    if (tid < K_DIM) vbuf[tid] = v[tid];
    __syncthreads();

    // cooperatively materialize s = exp(logits + u + v) (0 beyond N);
    // prefetch the x rows this block will stream in the WMMA loop.
    for (int i = 0; i < POOL_CHUNK / 128; ++i) {
        const int r = tid + 128 * i;
        const int n = base + r;
        const int nc = (n < N) ? n : (N - 1);
        __builtin_prefetch(x + (size_t)nc * C_DIM, 0, 0);  // global_prefetch_b8
        if (n < N) {
            const float un = u[n];
            const float4* lr = reinterpret_cast<const float4*>(logits + (size_t)n * K_DIM);
            #pragma unroll
            for (int q = 0; q < 8; ++q) {
                float4 t = lr[q];
                sbuf[r * K_DIM + 4 * q + 0] = __expf(t.x + un + vbuf[4 * q + 0]);
                sbuf[r * K_DIM + 4 * q + 1] = __expf(t.y + un + vbuf[4 * q + 1]);
                sbuf[r * K_DIM + 4 * q + 2] = __expf(t.z + un + vbuf[4 * q + 2]);
                sbuf[r * K_DIM + 4 * q + 3] = __expf(t.w + un + vbuf[4 * q + 3]);
            }
            bbuf[r] = batch[n];
        } else {
            #pragma unroll
            for (int k = 0; k < K_DIM; ++k) sbuf[r * K_DIM + k] = 0.f;
            bbuf[r] = batch[N - 1];   // harmless: s == 0
        }
    }
    __syncthreads();

    const int bLo = bbuf[0];
    const int bHi = bbuf[POOL_CHUNK - 1];

    for (int b = bLo; b <= bHi; ++b) {
        v8f acc00 = vzero8(), acc01 = vzero8();  // kt0 x {ct0, ct1}
        v8f acc10 = vzero8(), acc11 = vzero8();  // kt1 x {ct0, ct1}

        for (int n0 = 0; n0 < POOL_CHUNK; n0 += 4) {
            const int nA = n0 + half * 2;        // reduction rows for this half
            const bool m0 = (bbuf[nA]     == b);
            const bool m1 = (bbuf[nA + 1] == b);
            v2f a0, a1;
            a0.x = m0 ? sbuf[nA * K_DIM + l16]            : 0.f;
            a0.y = m1 ? sbuf[(nA + 1) * K_DIM + l16]      : 0.f;
            a1.x = m0 ? sbuf[nA * K_DIM + 16 + l16]       : 0.f;
            a1.y = m1 ? sbuf[(nA + 1) * K_DIM + 16 + l16] : 0.f;

            const int g0 = min(base + nA,     N - 1);
            const int g1 = min(base + nA + 1, N - 1);
            const int c0 = (2 * wav + 0) * 16 + l16;
            const int c1 = (2 * wav + 1) * 16 + l16;
            v2f bx0, bx1;
            bx0.x = x[(size_t)g0 * C_DIM + c0];
            bx0.y = x[(size_t)g1 * C_DIM + c0];
            bx1.x = x[(size_t)g0 * C_DIM + c1];
            bx1.y = x[(size_t)g1 * C_DIM + c1];

            acc00 = wmma_f32_4(a0, bx0, acc00);
            acc01 = wmma_f32_4(a0, bx1, acc01);
            acc10 = wmma_f32_4(a1, bx0, acc10);
            acc11 = wmma_f32_4(a1, bx1, acc11);
        }

        float* ob = out + (size_t)b * K_DIM * C_DIM;
        #pragma unroll
        for (int j = 0; j < 8; ++j) {
            const int kr0 = 0  + j + half * 8;
            const int kr1 = 16 + j + half * 8;
            const int c0  = (2 * wav + 0) * 16 + l16;
            const int c1  = (2 * wav + 1) * 16 + l16;
            atomicAdd(&ob[kr0 * C_DIM + c0], acc00[j]);
            atomicAdd(&ob[kr0 * C_DIM + c1], acc01[j]);
            atomicAdd(&ob[kr1 * C_DIM + c0], acc10[j]);
            atomicAdd(&ob[kr1 * C_DIM + c1], acc11[j]);
        }
    }
}

// ---------------------------------------------------------------------------
extern "C" void kernel_launch(void* const* d_in, const int* in_sizes, int n_in,
                              void* d_out, int out_size, void* d_ws, size_t ws_size,
                              hipStream_t stream) {
    const float* x    = (const float*)d_in[0];
    const int*   bat  = (const int*)  d_in[1];
    const float* W1   = (const float*)d_in[2];
    const float* b1   = (const float*)d_in[3];
    const float* W2   = (const float*)d_in[4];
    const float* b2   = (const float*)d_in[5];
    const float* scal = (const float*)d_in[6];
    float*       out  = (float*)d_out;

    const int N = in_sizes[0] / C_DIM;   // 200000

    // workspace layout (floats)
    float* ws      = (float*)d_ws;
    float* logits  = ws;                                 // N*32
    float* u       = logits + (size_t)N * K_DIM;         // N
    float* v       = u + N;                              // 32
    float* pmax    = v + K_DIM;                          // nblk*32
    const int nblk = (N + 255) / 256;
    float* psum    = pmax + (size_t)nblk * K_DIM;        // nblk*32
    (void)ws_size; (void)n_in;

    const float logT = logf((float)N / (float)K_DIM + 1e-9f);

    // 0) zero out + v
    k_init<<<(out_size + 255) / 256, 256, 0, stream>>>(out, out_size, v);

    // 1) fused MLP -> logits (TDM staging + WMMA f32)
    k_logits<<<(N + 63) / 64, 128, 0, stream>>>(x, W1, b1, W2, b2, scal, logits, N);

    // 2) 20 Sinkhorn iterations (L2-resident sweeps)
    for (int it = 0; it < 20; ++it) {
        k_sinkrow<<<nblk, 256, 0, stream>>>(logits, v, u, pmax, psum, N);
        k_sinkcol<<<1, 32, 0, stream>>>(pmax, psum, v, nblk, logT);
    }

    // 3) batch-masked pooling GEMM (WMMA f32) with atomic scatter
    k_pool<<<(N + POOL_CHUNK - 1) / POOL_CHUNK, 128, 0, stream>>>(
        x, bat, logits, u, v, out, N);
}